// GCNLayer_14620068675903
// MI455X (gfx1250) — compile-verified
//
#include <hip/hip_runtime.h>
#include <stdint.h>

#define D_FEAT 64
#define EPW 128                // edges per wave (chunk)
#define WAVES_PER_BLOCK 8
#define BLOCK_THREADS (WAVES_PER_BLOCK * 32)

// ---------- zero-fill output (segment_sum semantics: empty rows = 0) ----------
__global__ void zero_out_kernel(float* __restrict__ p, int n) {
    int i = (blockIdx.x * blockDim.x + threadIdx.x) * 4;
    if (i + 3 < n) {
        float4 z; z.x = 0.f; z.y = 0.f; z.z = 0.f; z.w = 0.f;
        *reinterpret_cast<float4*>(p + i) = z;
    } else {
        for (; i < n; ++i) p[i] = 0.f;
    }
}

// ---------- CDNA5 async copy: global -> LDS, 16B per lane ----------
__device__ __forceinline__ void async_copy_b128(uint32_t lds_addr, const void* gptr) {
    asm volatile("global_load_async_to_lds_b128 %0, %1, off"
                 :: "v"(lds_addr), "v"((uint64_t)(uintptr_t)gptr)
                 : "memory");
}
__device__ __forceinline__ void wait_asynccnt0() {
    asm volatile("s_wait_asynccnt 0x0" ::: "memory");
}

// ---------- LDS group read: 3 x ds_load_b32 off one address VGPR ----------
// rows at +0, cols at +512, vals at +1024 within the wave's LDS block.
__device__ __forceinline__ void lds_read3(uint32_t addr, int& r, int& c, float& v) {
    asm volatile("ds_load_b32 %0, %3\n\t"
                 "ds_load_b32 %1, %3 offset:512\n\t"
                 "ds_load_b32 %2, %3 offset:1024\n\t"
                 "s_wait_dscnt 0x0"
                 : "=&v"(r), "=&v"(c), "=&v"(v)
                 : "v"(addr)
                 : "memory");
}

// ---------- helpers ----------
__device__ __forceinline__ float readlane_f(float x, int l) {
    return __int_as_float(__builtin_amdgcn_readlane(__float_as_int(x), l));
}

__device__ __forceinline__ void flush_row(float* __restrict__ out, int row, int lane,
                                          float ax, float ay, bool use_atomic) {
    float* p = out + (size_t)row * D_FEAT + lane * 2;
    if (use_atomic) {
        atomicAdd(p, ax);          // global_atomic_add_f32
        atomicAdd(p + 1, ay);
    } else {
        float2 v; v.x = ax; v.y = ay;
        *reinterpret_cast<float2*>(p) = v;   // global_store_b64
    }
}

struct SegState {
    float ax, ay;
    int   cur;
    bool  touches_left;
};

// Consume one 32-edge group already resident in (r, c, v) lane registers.
__device__ __forceinline__ void consume_group(const float* __restrict__ embeds,
                                              float* __restrict__ out,
                                              int lane, int n,
                                              int r, int c, float v,
                                              int row_left, SegState& s) {
    for (int j = 0; j < n; ++j) {
        const int   rj = __builtin_amdgcn_readlane(r, j);
        const int   cj = __builtin_amdgcn_readlane(c, j);
        const float vj = readlane_f(v, j);

        if (rj != s.cur) {
            // segment ended strictly inside chunk unless it touches e0
            const bool at = s.touches_left && (s.cur == row_left);
            flush_row(out, s.cur, lane, s.ax, s.ay, at);
            s.ax = 0.f; s.ay = 0.f;
            s.cur = rj;
            s.touches_left = false;
        }

        // prefetch the gather row 8 edges ahead (global_prefetch_b8)
        if (j + 8 < n) {
            const int cp = __builtin_amdgcn_readlane(c, j + 8);
            __builtin_prefetch(embeds + (size_t)cp * D_FEAT + lane * 2, 0, 3);
        }

        // gather 256B row (coalesced global_load_b64 per lane) and accumulate
        const float2 f = *reinterpret_cast<const float2*>(
            embeds + (size_t)cj * D_FEAT + lane * 2);
        s.ax = fmaf(vj, f.x, s.ax);
        s.ay = fmaf(vj, f.y, s.ay);
    }
}

// ---------- segmented-reduction COO SpMM ----------
// One wave per EPW-edge chunk. Lane = feature pair (float2) -> wave covers a full
// 64-float row (256B, coalesced). Edge scalars staged into a per-wave LDS block
// by the async DMA path (global_load_async_to_lds_b128), read back with explicit
// ds_load_b32 (asm, immediate offsets), broadcast wave-wide via v_readlane_b32.
// rows[] sorted => plain stores for interior segments, atomics only at chunk
// boundaries shared with a neighbor chunk.
__global__ void __launch_bounds__(BLOCK_THREADS)
spmm_coo_segred(const float* __restrict__ embeds,
                const int*   __restrict__ rows,
                const int*   __restrict__ cols,
                const float* __restrict__ vals,
                float* __restrict__ out, int E) {
    // per-wave block: [0,512) rows, [512,1024) cols, [1024,1536) vals (bytes)
    __shared__ int sh_edges[WAVES_PER_BLOCK][3 * EPW];

    const int lane = threadIdx.x & 31;
    const int w    = threadIdx.x >> 5;
    const int wave = blockIdx.x * WAVES_PER_BLOCK + w;
    const int e0 = wave * EPW;
    if (e0 >= E) return;
    const int e1 = min(e0 + EPW, E);
    const bool full = (e1 - e0) == EPW;

    const uint32_t lds_base = (uint32_t)(size_t)&sh_edges[w][0];

    // Stage this wave's whole edge chunk into LDS asynchronously:
    // 3 x b128 async ops, each lane moves 16B (512B per array per wave).
    if (full) {
        const int boff = lane * 16;
        async_copy_b128(lds_base +    0 + boff, (const char*)(rows + e0) + boff);
        async_copy_b128(lds_base +  512 + boff, (const char*)(cols + e0) + boff);
        async_copy_b128(lds_base + 1024 + boff, (const char*)(vals + e0) + boff);
    }

    // boundary rows (uniform scalar loads, overlap with async copy)
    const int row_left  = (e0 > 0) ? rows[e0 - 1] : -1;
    const int row_right = (e1 < E) ? rows[e1]     : -1;

    SegState s;
    s.ax = 0.f; s.ay = 0.f;
    s.cur = rows[e0];
    s.touches_left = true;   // current segment includes edge e0

    if (full) {
        wait_asynccnt0();
        #pragma unroll 1
        for (int base = 0; base < EPW; base += 32) {
            int r, c; float v;
            lds_read3(lds_base + (uint32_t)(base + lane) * 4u, r, c, v);
            consume_group(embeds, out, lane, 32, r, c, v, row_left, s);
        }
    } else {
        #pragma unroll 1
        for (int base = e0; base < e1; base += 32) {
            const int n   = min(32, e1 - base);
            const int idx = min(base + lane, E - 1);
            const int   r = rows[idx];
            const int   c = cols[idx];
            const float v = vals[idx];
            consume_group(embeds, out, lane, n, r, c, v, row_left, s);
        }
    }

    // final flush: may touch left and/or right chunk boundary
    const bool at = (s.touches_left && (s.cur == row_left)) || (s.cur == row_right);
    flush_row(out, s.cur, lane, s.ax, s.ay, at);
}

extern "C" void kernel_launch(void* const* d_in, const int* in_sizes, int n_in,
                              void* d_out, int out_size, void* d_ws, size_t ws_size,
                              hipStream_t stream) {
    const float* embeds = (const float*)d_in[0];
    const int*   rows   = (const int*)d_in[1];
    const int*   cols   = (const int*)d_in[2];
    const float* vals   = (const float*)d_in[3];
    float* out = (float*)d_out;

    const int E = in_sizes[1];

    // 1) zero output (segment_sum of empty segments = 0; atomics accumulate onto 0)
    {
        const int n4 = (out_size + 3) / 4;
        const int blk = 256;
        zero_out_kernel<<<(n4 + blk - 1) / blk, blk, 0, stream>>>(out, out_size);
    }

    // 2) segmented-reduction SpMM
    {
        const int nWaves  = (E + EPW - 1) / EPW;
        const int nBlocks = (nWaves + WAVES_PER_BLOCK - 1) / WAVES_PER_BLOCK;
        spmm_coo_segred<<<nBlocks, BLOCK_THREADS, 0, stream>>>(
            embeds, rows, cols, vals, out, E);
    }
}